// SDMAttention_22909355556999
// MI455X (gfx1250) — compile-verified
//
#include <hip/hip_runtime.h>
#include <hip/hip_bf16.h>
#include <math.h>

#define BB   128
#define SS   48
#define DD   512
#define HALF 24

typedef __attribute__((ext_vector_type(2))) float v2f;
typedef __attribute__((ext_vector_type(8))) float v8f;

// ---------------------------------------------------------------------------
// Kernel A: scores columns s = 24..47.
// One block per (s-24, b). 256 threads = 8 waves, 3 rows per wave.
// scores[b, l, s] = mean_d  exp(e_l) / (24 + sum_i exp(wn[i][d]))
//   where wn[i][:] = softmax_d(w_raw[b,i,s,:]),  e_l = 1e-9 (l<24) or wn[l-24][d]
// ---------------------------------------------------------------------------
__global__ __launch_bounds__(256) void k_scores_hi(const float* __restrict__ w_raw,
                                                   float* __restrict__ scores) {
    const int scol = blockIdx.x + HALF;   // 24..47
    const int b    = blockIdx.y;
    const int tid  = threadIdx.x;
    const int lane = tid & 31;
    const int wave = tid >> 5;            // 0..7

    __shared__ float E[HALF][DD];         // exp(softmax value), 48 KB
    __shared__ float r[DD];               // 1/denom per d
    __shared__ float red8[8];
    __shared__ float scLo;

    // phase 1: per-row softmax over d, then exp of it
    for (int i = wave * 3; i < wave * 3 + 3; ++i) {
        const float* row = w_raw + (((size_t)b * HALF + i) * SS + scol) * DD;
        float v[16];
        float mx = -INFINITY;
        #pragma unroll
        for (int e = 0; e < 16; ++e) {
            v[e] = row[lane * 16 + e];
            mx = fmaxf(mx, v[e]);
        }
        for (int off = 16; off > 0; off >>= 1)
            mx = fmaxf(mx, __shfl_xor(mx, off, 32));
        float se = 0.f;
        #pragma unroll
        for (int e = 0; e < 16; ++e) { v[e] = __expf(v[e] - mx); se += v[e]; }
        for (int off = 16; off > 0; off >>= 1)
            se += __shfl_xor(se, off, 32);
        const float inv = 1.0f / se;
        #pragma unroll
        for (int e = 0; e < 16; ++e)
            E[i][lane * 16 + e] = __expf(v[e] * inv);
    }
    __syncthreads();

    // phase 2: r[d] = 1/(24*exp(1e-9) + sum_i E[i][d]); exp(1e-9)==1 in f32
    float pr = 0.f;
    for (int d = tid; d < DD; d += 256) {
        float s = 24.0f;
        #pragma unroll
        for (int i = 0; i < HALF; ++i) s += E[i][d];
        const float rv = 1.0f / s;
        r[d] = rv;
        pr += rv;
    }
    for (int off = 16; off > 0; off >>= 1)
        pr += __shfl_xor(pr, off, 32);
    if (lane == 0) red8[wave] = pr;
    __syncthreads();
    if (tid == 0) {
        float t = 0.f;
        for (int w = 0; w < 8; ++w) t += red8[w];
        scLo = t * (1.0f / (float)DD);
    }
    __syncthreads();

    // phase 3: hi rows: mean_d E[i][d]*r[d]
    for (int i = wave * 3; i < wave * 3 + 3; ++i) {
        float s = 0.f;
        #pragma unroll
        for (int e = 0; e < 16; ++e) {
            const int d = lane * 16 + e;
            s += E[i][d] * r[d];
        }
        for (int off = 16; off > 0; off >>= 1)
            s += __shfl_xor(s, off, 32);
        if (lane == 0)
            scores[((size_t)b * SS + HALF + i) * SS + scol] = s * (1.0f / (float)DD);
    }
    if (tid < HALF)
        scores[((size_t)b * SS + tid) * SS + scol] = scLo;
}

// ---------------------------------------------------------------------------
// Kernel B: scalar recurrence per batch -> cs[24] -> scores columns s = 0..23.
// One block per b, 256 threads.
// ---------------------------------------------------------------------------
__global__ __launch_bounds__(256) void k_recur(const float* __restrict__ values,
                                               const float* __restrict__ u0,
                                               const float* __restrict__ w_raw,
                                               float* __restrict__ scores) {
    const int b    = blockIdx.x;
    const int tid  = threadIdx.x;
    const int lane = tid & 31;
    const int wave = tid >> 5;

    __shared__ float dls[HALF][DD];       // d = y - x, 48 KB
    __shared__ float red8[8];
    __shared__ float rowsum_u0[HALF];
    __shared__ float slotdot[HALF];
    __shared__ float Dtot_sh;
    __shared__ float cs[HALF];

    // build d rows; accumulate Dtot
    float part = 0.f;
    for (int idx = tid; idx < HALF * DD; idx += 256) {
        const int j  = idx >> 9;
        const int dd = idx & (DD - 1);
        const float x = values[((size_t)b * SS + j) * DD + dd];
        const float y = values[((size_t)b * SS + HALF + j) * DD + dd];
        const float dv = y - x;
        dls[j][dd] = dv;
        part += dv;
    }
    for (int off = 16; off > 0; off >>= 1)
        part += __shfl_xor(part, off, 32);
    if (lane == 0) red8[wave] = part;
    __syncthreads();
    if (tid == 0) {
        float s = 0.f;
        for (int w = 0; w < 8; ++w) s += red8[w];
        Dtot_sh = s;
    }
    __syncthreads();

    // rowsum_u0[j] = sum_d u0[b,j,0,:]
    for (int j = wave * 3; j < wave * 3 + 3; ++j) {
        const float* row = u0 + ((size_t)b * HALF + j) * DD;
        float s = 0.f;
        for (int e = lane; e < DD; e += 32) s += row[e];
        for (int off = 16; off > 0; off >>= 1)
            s += __shfl_xor(s, off, 32);
        if (lane == 0) rowsum_u0[j] = s;
    }

    // slotdot[j] = dot(softmax_d(w_raw[b,j,47,:]), d[j,:])
    for (int j = wave * 3; j < wave * 3 + 3; ++j) {
        const float* row = w_raw + (((size_t)b * HALF + j) * SS + (SS - 1)) * DD;
        float mx = -INFINITY;
        for (int e = lane; e < DD; e += 32) mx = fmaxf(mx, row[e]);
        for (int off = 16; off > 0; off >>= 1)
            mx = fmaxf(mx, __shfl_xor(mx, off, 32));
        float se = 0.f, dot = 0.f;
        for (int e = lane; e < DD; e += 32) {
            const float ex = __expf(row[e] - mx);
            se += ex;
            dot += ex * dls[j][e];
        }
        for (int off = 16; off > 0; off >>= 1) {
            se  += __shfl_xor(se, off, 32);
            dot += __shfl_xor(dot, off, 32);
        }
        if (lane == 0) slotdot[j] = dot / se;
    }
    __syncthreads();

    // sequential 24-step recurrence (scalar per batch)
    if (tid == 0) {
        float su0 = 0.f, sd = 0.f;
        for (int j = 0; j < HALF; ++j) { su0 += rowsum_u0[j]; sd += slotdot[j]; }
        float total = su0;     // c0 uses the ORIGINAL u0 (before slot-0 replacement)
        float cprev = 0.f;
        for (int s = 0; s < HALF; ++s) {
            float lamb = total * (1.0f / (float)(HALF * DD));
            lamb *= lamb;
            float mx = -INFINITY;
            for (int j = 0; j < HALF; ++j) mx = fmaxf(mx, -lamb * dls[j][s]);
            float Z = 0.f;
            for (int j = 0; j < HALF; ++j) Z += __expf(-lamb * dls[j][s] - mx);
            const float c = __expf(-lamb * dls[HALF - 1][s] - mx) / Z;
            cs[s] = c;
            if (s == 0) total += -rowsum_u0[0] + sd;                 // slot 0 <- w_init-weighted sum
            else        total += -rowsum_u0[s] + cprev * Dtot_sh;    // slot s <- c_{s-1} * d_sum
            cprev = c;
        }
    }
    __syncthreads();

    // columns s < 24: softmax over {24x 1e-9, 24x cs[s]}; constant over d so mean == value
    for (int idx = tid; idx < SS * HALF; idx += 256) {
        const int l = idx / HALF;
        const int s = idx % HALF;
        const float ec  = __expf(cs[s]);
        const float inv = 1.0f / (24.0f + 24.0f * ec);   // exp(1e-9)==1 in f32
        scores[((size_t)b * SS + l) * SS + s] = (l < HALF) ? inv : ec * inv;
    }
}

// ---------------------------------------------------------------------------
// Kernel C: V[b] = scores[b] (48x48) x values[b] (48x512) via V_WMMA_F32_16X16X4_F32.
// One wave per 16x16 output tile; 12 WMMAs per tile (K=48).
// A 16x4 layout: lanes 0-15 -> K0,K1 ; lanes 16-31 -> K2,K3 (row = m0+lane&15)
// B 4x16 layout: symmetric with N in place of M.
// C/D: VGPR r -> row m0+r (lanes 0-15) / m0+8+r (lanes 16-31), col n0+(lane&15).
// ---------------------------------------------------------------------------
__global__ __launch_bounds__(32) void k_gemm(const float* __restrict__ scores,
                                             const float* __restrict__ values,
                                             float* __restrict__ out) {
    const int b    = blockIdx.z;
    const int m0   = blockIdx.y * 16;
    const int n0   = blockIdx.x * 16;
    const int lane = threadIdx.x;
    const int half = lane >> 4;       // 0 or 1
    const int l15  = lane & 15;

    const float* A  = scores + (size_t)b * SS * SS;   // 48x48 row-major
    const float* Bm = values + (size_t)b * SS * DD;   // 48x512 row-major
    v8f c = {};
    #pragma unroll
    for (int k0 = 0; k0 < SS; k0 += 4) {
        v2f a, bf;
        a.x  = A[(m0 + l15) * SS + k0 + 2 * half + 0];
        a.y  = A[(m0 + l15) * SS + k0 + 2 * half + 1];
        bf.x = Bm[(k0 + 2 * half + 0) * DD + n0 + l15];
        bf.y = Bm[(k0 + 2 * half + 1) * DD + n0 + l15];
        c = __builtin_amdgcn_wmma_f32_16x16x4_f32(false, a, false, bf,
                                                  (short)0, c, false, false);
    }
    float* O = out + (size_t)b * SS * DD;
    #pragma unroll
    for (int r = 0; r < 8; ++r)
        O[(m0 + r + 8 * half) * DD + n0 + l15] = c[r];
}

// ---------------------------------------------------------------------------
extern "C" void kernel_launch(void* const* d_in, const int* in_sizes, int n_in,
                              void* d_out, int out_size, void* d_ws, size_t ws_size,
                              hipStream_t stream) {
    (void)in_sizes; (void)n_in; (void)out_size; (void)ws_size;
    // inputs: 0=queries (unused), 1=keys (unused), 2=values, 3=attn_mask (unused),
    //         4=u0, 5=w_raw
    const float* values = (const float*)d_in[2];
    const float* u0     = (const float*)d_in[4];
    const float* w_raw  = (const float*)d_in[5];
    float* scores = (float*)d_ws;            // B*48*48 f32 = 1.18 MB
    float* out    = (float*)d_out;           // (128,48,1,512) f32

    dim3 gA(HALF, BB);                       // 24 x 128 blocks
    k_scores_hi<<<gA, 256, 0, stream>>>(w_raw, scores);

    k_recur<<<BB, 256, 0, stream>>>(values, u0, w_raw, scores);

    dim3 gC(DD / 16, SS / 16, BB);           // 32 x 3 x 128 tiles
    k_gemm<<<gC, 32, 0, stream>>>(scores, values, out);
}